// MultigridLayer_20959440405200
// MI455X (gfx1250) — compile-verified
//
#include <hip/hip_runtime.h>
#include <stdint.h>

typedef __attribute__((ext_vector_type(2))) float v2f;
typedef __attribute__((ext_vector_type(8))) float v8f;

#define NUM_VAR 12288
#define NUM_EPS 4096
#define NCOLS   (NUM_VAR + NUM_EPS)   // 16384
#define M_DIM   NUM_EPS               // 4096
#define PINV_LO 1e5f                  // 1/DS for n <  NUM_VAR
#define PINV_HI 1e-5f                 // 1/US for n >= NUM_VAR

__device__ __forceinline__ float pinv_of(int n) {
    return (n < NUM_VAR) ? PINV_LO : PINV_HI;
}

// ---------------------------------------------------------------------------
// Small outputs: downsampled coeffs/rhs (jax antialiased linear, factor 2 ->
// separable [1,3,3,1]/8 taps, edge-renormalized) and paired step sums.
// ---------------------------------------------------------------------------
__device__ __forceinline__ float bilin4(const float* __restrict__ p, int xs,
                                        int y, int x) {
    const float w[4] = {0.125f, 0.375f, 0.375f, 0.125f};
    float accy = 0.f, wsy = 0.f;
    #pragma unroll
    for (int t = 0; t < 4; ++t) {
        int iy = 2 * y - 1 + t;
        if (iy < 0 || iy > 63) continue;
        float accx = 0.f, wsx = 0.f;
        #pragma unroll
        for (int u = 0; u < 4; ++u) {
            int ix = 2 * x - 1 + u;
            if (ix < 0 || ix > 63) continue;
            accx += w[u] * p[(iy * 64 + ix) * xs];
            wsx  += w[u];
        }
        accy += w[t] * (accx / wsx);
        wsy  += w[t];
    }
    return accy / wsy;
}

__global__ void small_outputs_kernel(const float* __restrict__ coeffs,
                                     const float* __restrict__ rhs,
                                     const float* __restrict__ steps_x,
                                     const float* __restrict__ steps_y,
                                     float* __restrict__ out) {
    const int C_N = 2 * 1024 * 6;   // 12288
    const int R_N = 2 * 1024;       // 2048
    const int S_N = 2 * 31;         // 62
    int gidx = blockIdx.x * blockDim.x + threadIdx.x;
    int idx = gidx;
    if (idx < C_N) {
        int b = idx / 6144, rem = idx % 6144;
        int p = rem / 6, o = rem % 6;
        int y = p / 32, x = p % 32;
        const float* base = coeffs + (size_t)b * 4096 * 6 + o;   // stride 6 per grid pt
        out[gidx] = bilin4(base, 6, y, x);
        return;
    }
    idx -= C_N;
    if (idx < R_N) {
        int b = idx / 1024, p = idx % 1024;
        int y = p / 32, x = p % 32;
        out[gidx] = bilin4(rhs + (size_t)b * 4096, 1, y, x);
        return;
    }
    idx -= R_N;
    if (idx < S_N) {
        int b = idx / 31, i = idx % 31;
        out[gidx] = steps_x[b * 63 + 2 * i] + steps_x[b * 63 + 2 * i + 1];
        return;
    }
    idx -= S_N;
    if (idx < S_N) {
        int b = idx / 31, i = idx % 31;
        out[gidx] = steps_y[b * 63 + 2 * i] + steps_y[b * 63 + 2 * i + 1];
    }
}

// ---------------------------------------------------------------------------
// D[b,m] = sum_n Pinv[n] * A[b,m,n]^2   -- one block per (b,m) row, float4 loads
// ---------------------------------------------------------------------------
__global__ __launch_bounds__(256) void d_diag_kernel(const float* __restrict__ A,
                                                     float* __restrict__ D) {
    const int bm = blockIdx.x;                     // 0..8191
    const float4* r4 = (const float4*)(A + (size_t)bm * NCOLS);
    const int tid = threadIdx.x;
    float s = 0.f;
    for (int i = tid; i < NCOLS / 4; i += 256) {
        float4 v = r4[i];
        int n = i * 4;
        s += v.x * v.x * pinv_of(n)     + v.y * v.y * pinv_of(n + 1)
           + v.z * v.z * pinv_of(n + 2) + v.w * v.w * pinv_of(n + 3);
    }
    __shared__ float red[256];
    red[tid] = s;
    __syncthreads();
    for (int off = 128; off > 0; off >>= 1) {
        if (tid < off) red[tid] += red[tid + off];
        __syncthreads();
    }
    if (tid == 0) D[bm] = red[0];
}

// ---------------------------------------------------------------------------
// AAt[m,p] = sum_n A0[m,n] * Pinv[n] * A0[p,n]
// 128x128 output tile / block, K-panel 32, fp32 WMMA 16x16x4,
// double-buffered GLOBAL_LOAD_ASYNC_TO_LDS_B128 panel staging (ASYNCcnt).
// Pinv is panel-constant (12288 % 32 == 0): accumulate raw, rescale acc by
// 1e10 at the region boundary, scale by 1e-5 at store.
// ---------------------------------------------------------------------------
#define BM 128
#define BN 128
#define BK 32
#define LDP (BK + 2)                 // padded LDS row stride (floats)
#define PANEL ((BM + BN) * LDP)      // floats per buffer

__device__ __forceinline__ void async_ld_b128(uint32_t lds_byte_off,
                                              const float* g) {
    asm volatile("global_load_async_to_lds_b128 %0, %1, off"
                 :: "v"(lds_byte_off),
                    "v"((unsigned long long)(uintptr_t)g)
                 : "memory");
}

__global__ __launch_bounds__(256) void aat_wmma_kernel(const float* __restrict__ A0,
                                                       float* __restrict__ AAt) {
    __shared__ float lds[2 * PANEL];
    const int tid  = threadIdx.x;
    const int lane = tid & 31;
    const int wave = tid >> 5;          // 0..7
    const int wy = wave >> 1;           // 0..3 (M)
    const int wx = wave & 1;            // 0..1 (N)
    const int m_base = blockIdx.y * BM;
    const int n_base = blockIdx.x * BN;
    const int half = lane >> 4;         // 0 | 1
    const int hr   = lane & 15;         // 0..15

    const uint32_t lds_base = (uint32_t)(uintptr_t)(void*)lds;  // flat LDS: addr[31:0]

    v8f acc[2][4];
    #pragma unroll
    for (int i = 0; i < 2; ++i)
        #pragma unroll
        for (int j = 0; j < 4; ++j)
            acc[i][j] = v8f{};

    // issue one panel (both sides): 8 async b128 per thread, FIFO-ordered
    auto issue_panel = [&](int buf, int k0) {
        const uint32_t bufA = lds_base + (uint32_t)(buf * PANEL) * 4u;
        const uint32_t bufB = bufA + (uint32_t)(BM * LDP) * 4u;
        #pragma unroll
        for (int it = 0; it < 4; ++it) {
            int idx = tid + it * 256;             // 0..1023
            int row = idx >> 3;
            int c   = (idx & 7) * 4;
            uint32_t loff = (uint32_t)((row * LDP + c) * 4);
            async_ld_b128(bufA + loff, A0 + (size_t)(m_base + row) * NCOLS + k0 + c);
            async_ld_b128(bufB + loff, A0 + (size_t)(n_base + row) * NCOLS + k0 + c);
        }
    };

    issue_panel(0, 0);
    int buf = 0;
    for (int k0 = 0; k0 < NCOLS; k0 += BK) {
        const bool has_next = (k0 + BK) < NCOLS;
        if (has_next) {
            issue_panel(buf ^ 1, k0 + BK);
            // 8 next-panel copies may stay in flight; current panel is done
            asm volatile("s_wait_asynccnt 8" ::: "memory");
        } else {
            asm volatile("s_wait_asynccnt 0" ::: "memory");
        }
        __syncthreads();

        if (k0 == NUM_VAR) {   // region boundary: acc = S1 -> 1e10*S1
            #pragma unroll
            for (int i = 0; i < 2; ++i)
                #pragma unroll
                for (int j = 0; j < 4; ++j)
                    #pragma unroll
                    for (int v = 0; v < 8; ++v)
                        acc[i][j][v] *= 1e10f;
        }

        const float* sA = lds + buf * PANEL;
        const float* sB = sA + BM * LDP;
        #pragma unroll
        for (int kk = 0; kk < BK; kk += 4) {
            const int ks = kk + (half << 1);   // lane<16 -> K=0,1; lane>=16 -> K=2,3
            v2f af[2], bf[4];
            #pragma unroll
            for (int i = 0; i < 2; ++i) {
                const float* p = &sA[(wy * 32 + i * 16 + hr) * LDP + ks];
                af[i].x = p[0]; af[i].y = p[1];
            }
            #pragma unroll
            for (int j = 0; j < 4; ++j) {
                const float* p = &sB[(wx * 64 + j * 16 + hr) * LDP + ks];
                bf[j].x = p[0]; bf[j].y = p[1];
            }
            #pragma unroll
            for (int i = 0; i < 2; ++i)
                #pragma unroll
                for (int j = 0; j < 4; ++j)
                    acc[i][j] = __builtin_amdgcn_wmma_f32_16x16x4_f32(
                        false, af[i], false, bf[j],
                        (short)0, acc[i][j], false, false);
        }
        __syncthreads();
        buf ^= 1;
    }

    // C/D layout: VGPR v -> row (half*8 + v), col = hr; fold final 1e-5 scale in
    #pragma unroll
    for (int i = 0; i < 2; ++i) {
        #pragma unroll
        for (int j = 0; j < 4; ++j) {
            const int col = n_base + wx * 64 + j * 16 + hr;
            #pragma unroll
            for (int v = 0; v < 8; ++v) {
                const int row = m_base + wy * 32 + i * 16 + half * 8 + v;
                AAt[(size_t)row * M_DIM + col] = acc[i][j][v] * PINV_HI;
            }
        }
    }
}

// ---------------------------------------------------------------------------
extern "C" void kernel_launch(void* const* d_in, const int* in_sizes, int n_in,
                              void* d_out, int out_size, void* d_ws, size_t ws_size,
                              hipStream_t stream) {
    const float* coeffs = (const float*)d_in[0];
    const float* rhs    = (const float*)d_in[1];
    const float* sx     = (const float*)d_in[2];
    const float* sy     = (const float*)d_in[3];
    const float* A      = (const float*)d_in[4];
    float* out = (float*)d_out;

    const size_t AAT_OFF = 14460;
    const size_t D_OFF   = AAT_OFF + (size_t)M_DIM * M_DIM;

    const int small_n = 14460;
    small_outputs_kernel<<<(small_n + 255) / 256, 256, 0, stream>>>(
        coeffs, rhs, sx, sy, out);

    d_diag_kernel<<<2 * M_DIM, 256, 0, stream>>>(A, out + D_OFF);

    dim3 grid(M_DIM / BN, M_DIM / BM);   // 32 x 32
    aat_wmma_kernel<<<grid, 256, 0, stream>>>(A, out + AAT_OFF);
}